// ROIAlign_39805756899535
// MI455X (gfx1250) — compile-verified
//
#include <hip/hip_runtime.h>
#include <stdint.h>

typedef __attribute__((ext_vector_type(2))) float v2f;
typedef __attribute__((ext_vector_type(8))) float v8f;

#define CROP   12
#define HW     160
#define NCH    256
#define NCP    128   // channel pairs
#define KCOLS  24    // 2 columns (x0,x1) per x-sample, 12 x-samples

// One block per box, 256 threads = 8 wave32.
// Per (box,y):  async-stage 2 rows x 24 cols x 128 channel-pairs (float2) into LDS,
// then 8 waves each do a 16x16 (M=wx pad, N=cp tile) x K=24 f32 WMMA GEMM that fuses
// x-interpolation + 2x2 (width-pair, channel-pair) average pooling.
__global__ __launch_bounds__(256)
void roialign_wmma_kernel(const float* __restrict__ feat,
                          const float* __restrict__ boxes,
                          float* __restrict__ out)
{
    __shared__ __align__(16) float2 raw[2][KCOLS][NCP];  // [y0/y1 row][j][cp] : 48 KB
    __shared__ __align__(16) float  Wp[16][KCOLS];       // A-matrix (rows 6..15 zero)
    __shared__ float wy0s[CROP], wy1s[CROP];
    __shared__ int   rowY0[CROP], rowY1[CROP], srcX[KCOLS];

    const int n   = blockIdx.x;
    const int tid = threadIdx.x;

    // ---- per-box precompute --------------------------------------------
    for (int e = tid; e < 16 * KCOLS; e += 256) (&Wp[0][0])[e] = 0.0f;

    float wx0r = 0.0f, wx1r = 0.0f;
    if (tid < CROP) {
        const float y1b = boxes[4 * n + 0], x1b = boxes[4 * n + 1];
        const float y2b = boxes[4 * n + 2], x2b = boxes[4 * n + 3];
        const float S = (float)(HW - 1);
        const float fi = (float)tid;
        const float ys = y1b * S + fi * ((y2b - y1b) * S / (float)(CROP - 1));
        const float xs = x1b * S + fi * ((x2b - x1b) * S / (float)(CROP - 1));
        const float vy = (ys >= 0.0f && ys <= S) ? 1.0f : 0.0f;
        const float vx = (xs >= 0.0f && xs <= S) ? 1.0f : 0.0f;
        const float y0f = floorf(ys), x0f = floorf(xs);
        const float ly = ys - y0f, lx = xs - x0f;
        rowY0[tid] = min(max((int)y0f, 0), HW - 1);
        rowY1[tid] = min(max((int)y0f + 1, 0), HW - 1);
        srcX[2 * tid]     = min(max((int)x0f, 0), HW - 1);
        srcX[2 * tid + 1] = min(max((int)x0f + 1, 0), HW - 1);
        wy0s[tid] = (1.0f - ly) * vy;
        wy1s[tid] = ly * vy;
        wx0r = (1.0f - lx) * vx * 0.25f;   // 0.25 = 2x2 pooling mean
        wx1r = lx * vx * 0.25f;
    }
    __syncthreads();
    if (tid < CROP) {                       // scatter A-matrix weights
        Wp[tid >> 1][2 * tid]     = wx0r;
        Wp[tid >> 1][2 * tid + 1] = wx1r;
    }
    __syncthreads();

    const uint32_t rawBase = (uint32_t)(uintptr_t)(&raw[0][0][0]);
    const int cp    = tid & (NCP - 1);      // fixed channel-pair per thread for staging
    const int wave  = tid >> 5;
    const int lane  = tid & 31;
    const int laneN = lane & 15;
    const int half  = lane >> 4;
    const int cpB   = wave * 16 + laneN;    // this lane's N (channel-pair) column

    for (int y = 0; y < CROP; ++y) {
        // ---- async gather: 2 rows x 24 cols x 128 cps of float2 into LDS ----
        const float* r0 = feat + (size_t)rowY0[y] * (HW * NCH) + 2 * cp;
        const float* r1 = feat + (size_t)rowY1[y] * (HW * NCH) + 2 * cp;
        int jj = (tid >> 7);                // 0 or 1, step 2 -> covers 0..47
        for (int r = 0; r < 24; ++r) {
            const int row = (jj >= KCOLS) ? 1 : 0;
            const int j   = jj - row * KCOLS;
            const float* src = (row ? r1 : r0) + (size_t)srcX[j] * NCH;
            const uint32_t lds = rawBase +
                (uint32_t)(((row * KCOLS + j) * NCP + cp) * sizeof(float2));
            asm volatile("global_load_async_to_lds_b64 %0, %1, off"
                         :: "v"(lds), "v"((uint64_t)(uintptr_t)src)
                         : "memory");
            jj += 2;
        }
        asm volatile("s_wait_asynccnt 0" ::: "memory");
        __syncthreads();

        // ---- WMMA: pooled[wx, cp] = sum_j Wp[wx,j] * h[j,cp] ----------------
        const float w0 = wy0s[y], w1 = wy1s[y];
        v8f acc = {};
        #pragma unroll
        for (int t = 0; t < 6; ++t) {
            const int kA = 4 * t + 2 * half;     // lanes 0-15: K=4t,4t+1 ; 16-31: 4t+2,4t+3
            v2f a = *(const v2f*)&Wp[laneN][kA];
            const float2 p0 = raw[0][kA][cpB],     q0 = raw[1][kA][cpB];
            const float2 p1 = raw[0][kA + 1][cpB], q1 = raw[1][kA + 1][cpB];
            v2f b;
            b.x = w0 * (p0.x + p0.y) + w1 * (q0.x + q0.y);  // h[kA,  cpB]
            b.y = w0 * (p1.x + p1.y) + w1 * (q1.x + q1.y);  // h[kA+1,cpB]
            acc = __builtin_amdgcn_wmma_f32_16x16x4_f32(
                      false, a, false, b, (short)0, acc, false, false);
        }

        // D layout: VGPR m, lanes 0-15 hold D[M=m][N=laneN]; M=0..5 are real wx rows.
        if (lane < 16) {
            #pragma unroll
            for (int m = 0; m < 6; ++m)
                out[(((size_t)n * 6 + m) * NCP + cpB) * CROP + y] = acc[m];
        }
        __syncthreads();   // protect raw[] before next iteration's async writes
    }
}

extern "C" void kernel_launch(void* const* d_in, const int* in_sizes, int n_in,
                              void* d_out, int out_size, void* d_ws, size_t ws_size,
                              hipStream_t stream) {
    const float* feat  = (const float*)d_in[0];   // [1,160,160,256] f32
    const float* boxes = (const float*)d_in[1];   // [2000,4] f32
    const int nBoxes = in_sizes[1] / 4;
    roialign_wmma_kernel<<<nBoxes, 256, 0, stream>>>(feat, boxes, (float*)d_out);
}